// ExponentialRepulsion_77120432767616
// MI455X (gfx1250) — compile-verified
//
#include <hip/hip_runtime.h>
#include <math.h>

#define BLOCK 256
#define TILE  1024            // edges per tile; TILE*12B dr + TILE*8B idx staged per tile
#define CHUNKS_DR 3           // 3 * 256 lanes * 16B = 12288B = TILE*12

__device__ __forceinline__ uint32_t lds_addr_of(const void* p) {
  // Generic pointers to LDS carry the LDS byte offset in the low 32 bits
  // (ISA 10.2: LDS aperture).
  return (uint32_t)(uintptr_t)p;
}

__global__ void zero_out_kernel(float* out) { out[0] = 0.0f; }

// Per-atom (A = |prefactor[Z]|, invR = 1/|scale[Z]|) table -> L2-resident 1.6MB.
__global__ void build_param_table(const int* __restrict__ Z,
                                  const float* __restrict__ rep_scale,
                                  const float* __restrict__ rep_pref,
                                  float2* __restrict__ table, int n_atoms) {
  int i = blockIdx.x * blockDim.x + threadIdx.x;
  if (i < n_atoms) {
    int z = Z[i];
    float A = fabsf(rep_pref[z]);
    float R = fabsf(rep_scale[z]);
    table[i] = make_float2(A, __builtin_amdgcn_rcpf(R));
  }
}

__device__ __forceinline__ float edge_energy(int ii, int jj,
                                             float x, float y, float z,
                                             float2 pli, float2 plj) {
  const float PI_OVER_RMAX = 1.57079632679489662f;  // pi / R_MAX, R_MAX = 2
  float d2 = fmaf(x, x, fmaf(y, y, z * z));
  float dr = __builtin_amdgcn_sqrtf(d2);
  dr = fminf(fmaxf(dr, 0.02f), 2.0f);
  float cosc = 0.5f * (__cosf(dr * PI_OVER_RMAX) + 1.0f);   // v_cos_f32
  float ex   = __expf(-dr * (pli.y + plj.y));               // v_exp_f32
  float num  = pli.x * plj.x * ex * cosc;
  float eij  = num * __builtin_amdgcn_rcpf(dr * dr);
  return (ii != jj) ? eij : 0.0f;
}

__global__ __launch_bounds__(BLOCK)
void edge_energy_kernel(const float* __restrict__ dr_vec,
                        const int*   __restrict__ idx,
                        const float2* __restrict__ params,
                        float* __restrict__ out,
                        int n_edges, int num_tiles) {
  __shared__ __align__(16) float s_dr[2][3 * TILE];   // 2 x 12288 B
  __shared__ __align__(16) int   s_idx[2][2 * TILE];  // 2 x  8192 B
  __shared__ float s_red[BLOCK];

  const int tid = threadIdx.x;
  const unsigned long long dr_base  = (unsigned long long)(uintptr_t)dr_vec;
  const unsigned long long idx_base = (unsigned long long)(uintptr_t)idx;
  float acc = 0.0f;

  // Issue one tile's async global->LDS copies (5 x B128 per lane).
  auto issue = [&](int t, int b) {
    uint32_t ebase = (uint32_t)t * (uint32_t)TILE;
    uint32_t gdr   = ebase * 12u;                 // byte offset into dr_vec
    uint32_t ldr   = lds_addr_of(&s_dr[b][0]);
#pragma unroll
    for (int c = 0; c < CHUNKS_DR; ++c) {
      uint32_t off = (uint32_t)(tid + c * BLOCK) * 16u;
      uint32_t l = ldr + off;
      uint32_t g = gdr + off;
      asm volatile("global_load_async_to_lds_b128 %0, %1, %2 th:TH_LOAD_NT"
                   :: "v"(l), "v"(g), "s"(dr_base) : "memory");
    }
    {
      uint32_t l = lds_addr_of(&s_idx[b][0]) + (uint32_t)tid * 16u;
      uint32_t g = ebase * 4u + (uint32_t)tid * 16u;
      asm volatile("global_load_async_to_lds_b128 %0, %1, %2 th:TH_LOAD_NT"
                   :: "v"(l), "v"(g), "s"(idx_base) : "memory");
    }
    {
      uint32_t l = lds_addr_of(&s_idx[b][TILE]) + (uint32_t)tid * 16u;
      uint32_t g = (uint32_t)n_edges * 4u + ebase * 4u + (uint32_t)tid * 16u;
      asm volatile("global_load_async_to_lds_b128 %0, %1, %2 th:TH_LOAD_NT"
                   :: "v"(l), "v"(g), "s"(idx_base) : "memory");
    }
  };

  int t = blockIdx.x;
  int p = 0;
  if (t < num_tiles) issue(t, 0);

  for (; t < num_tiles; t += gridDim.x) {
    int tn = t + gridDim.x;
    bool more = (tn < num_tiles);
    if (more) {
      issue(tn, p ^ 1);                                   // prefetch next tile
      asm volatile("s_wait_asynccnt 0x5" ::: "memory");   // current tile's 5 done
    } else {
      asm volatile("s_wait_asynccnt 0x0" ::: "memory");
    }
    __syncthreads();  // all waves' staged data visible

#pragma unroll
    for (int u = 0; u < TILE / BLOCK; ++u) {
      int e  = tid + u * BLOCK;                // conflict-free LDS banks in wave32
      int ii = s_idx[p][e];
      int jj = s_idx[p][TILE + e];
      float x = s_dr[p][3 * e + 0];
      float y = s_dr[p][3 * e + 1];
      float z = s_dr[p][3 * e + 2];
      float2 pli = params[ii];                 // L2-resident gather
      float2 plj = params[jj];
      acc += edge_energy(ii, jj, x, y, z, pli, plj);
    }
    __syncthreads();  // everyone done reading buf[p] before it is re-staged
    p ^= 1;
  }

  // Block reduction + one f32 atomic per block.
  s_red[tid] = acc;
  __syncthreads();
#pragma unroll
  for (int s = BLOCK / 2; s > 0; s >>= 1) {
    if (tid < s) s_red[tid] += s_red[tid + s];
    __syncthreads();
  }
  if (tid == 0) atomicAdd(out, s_red[0]);
}

// Generic tail (unused for N_EDGES = 12.8M which is a multiple of TILE).
__global__ __launch_bounds__(BLOCK)
void edge_tail_kernel(const float* __restrict__ dr_vec,
                      const int*   __restrict__ idx,
                      const float2* __restrict__ params,
                      float* __restrict__ out,
                      int start, int n_edges) {
  __shared__ float s_red[BLOCK];
  int tid = threadIdx.x;
  float acc = 0.0f;
  for (int e = start + blockIdx.x * BLOCK + tid; e < n_edges;
       e += gridDim.x * BLOCK) {
    int ii = idx[e];
    int jj = idx[n_edges + e];
    float x = dr_vec[3 * e + 0];
    float y = dr_vec[3 * e + 1];
    float z = dr_vec[3 * e + 2];
    acc += edge_energy(ii, jj, x, y, z, params[ii], params[jj]);
  }
  s_red[tid] = acc;
  __syncthreads();
  for (int s = BLOCK / 2; s > 0; s >>= 1) {
    if (tid < s) s_red[tid] += s_red[tid + s];
    __syncthreads();
  }
  if (tid == 0) atomicAdd(out, s_red[0]);
}

extern "C" void kernel_launch(void* const* d_in, const int* in_sizes, int n_in,
                              void* d_out, int out_size, void* d_ws, size_t ws_size,
                              hipStream_t stream) {
  // setup_inputs order: R, dr_vec, Z, idx, box, properties, rep_scale, rep_prefactor
  const float* dr_vec    = (const float*)d_in[1];
  const int*   Z         = (const int*)d_in[2];
  const int*   idx       = (const int*)d_in[3];
  const float* rep_scale = (const float*)d_in[6];
  const float* rep_pref  = (const float*)d_in[7];
  float* out = (float*)d_out;

  int n_atoms = in_sizes[2];
  int n_edges = in_sizes[3] / 2;     // idx is (2, N_EDGES)
  float2* table = (float2*)d_ws;     // n_atoms * 8 bytes

  zero_out_kernel<<<1, 1, 0, stream>>>(out);
  build_param_table<<<(n_atoms + BLOCK - 1) / BLOCK, BLOCK, 0, stream>>>(
      Z, rep_scale, rep_pref, table, n_atoms);

  int num_tiles = n_edges / TILE;    // 12500 for the reference sizes
  if (num_tiles > 0) {
    int grid = num_tiles < 2500 ? num_tiles : 2500;
    edge_energy_kernel<<<grid, BLOCK, 0, stream>>>(dr_vec, idx, table, out,
                                                   n_edges, num_tiles);
  }
  int rem = n_edges - num_tiles * TILE;
  if (rem > 0) {
    int g2 = (rem + BLOCK - 1) / BLOCK;
    edge_tail_kernel<<<g2, BLOCK, 0, stream>>>(dr_vec, idx, table, out,
                                               num_tiles * TILE, n_edges);
  }
}